// TGCNGraphConvolution_39367670235252
// MI455X (gfx1250) — compile-verified
//
#include <hip/hip_runtime.h>
#include <math.h>

#define NUM_HEADS 8
#define B_ 4
#define N_ 1024
#define D_ 65
#define OD 64

typedef __attribute__((ext_vector_type(2))) float v2f;
typedef __attribute__((ext_vector_type(8))) float v8f;

// workspace layout (float offsets), total = 655360 floats = 2.5 MB
#define WS_HT   0                               // (B,N,64) = h @ W
#define WS_HW   (WS_HT   + B_*N_*OD)            // (B,N,64) = h @ weights
#define WS_ESRC (WS_HW   + B_*N_*OD)            // (B,N,8)
#define WS_EDST (WS_ESRC + B_*N_*NUM_HEADS)     // (B,N,8)
#define WS_M    (WS_EDST + B_*N_*NUM_HEADS)     // (B,N,8) softmax row max
#define WS_S    (WS_M    + B_*N_*NUM_HEADS)     // (B,N,8) softmax row sum

// ---------------------------------------------------------------------------
// Kernel 1: per (b,n) row: h = [input, hidden]; ht = h@W; hw = h@weights;
//           e_src/e_dst attention logits. 4096 threads total.
// ---------------------------------------------------------------------------
__global__ void k_proj(const float* __restrict__ inp, const float* __restrict__ hid,
                       const float* __restrict__ W,   const float* __restrict__ attn,
                       const float* __restrict__ wts, float* __restrict__ ws) {
  int t = blockIdx.x * blockDim.x + threadIdx.x;   // 0..B*N-1
  float h[D_];
  h[0] = inp[t];
#pragma unroll
  for (int d = 0; d < 64; ++d) h[1 + d] = hid[t * 64 + d];

  float* ht = ws + WS_HT;
  float* hw = ws + WS_HW;

#pragma unroll
  for (int hh = 0; hh < NUM_HEADS; ++hh) {
    float es = 0.f, ed = 0.f;
    for (int f = 0; f < 8; ++f) {
      int k = hh * 8 + f;
      float a1 = 0.f, a2 = 0.f;
#pragma unroll
      for (int d = 0; d < D_; ++d) {
        a1 = fmaf(h[d], W[d * OD + k],   a1);   // uniform (scalar) weight loads
        a2 = fmaf(h[d], wts[d * OD + k], a2);
      }
      ht[t * OD + k] = a1;
      hw[t * OD + k] = a2;
      es = fmaf(a1, attn[f],     es);
      ed = fmaf(a1, attn[8 + f], ed);
    }
    ws[WS_ESRC + t * NUM_HEADS + hh] = es;
    ws[WS_EDST + t * NUM_HEADS + hh] = ed;
  }
}

// ---------------------------------------------------------------------------
// Kernel 2: streaming softmax stats per (b,i,h): row max m, row sum s.
// 32768 threads, each loops j = 0..1023. No 128MB attention tensor.
// ---------------------------------------------------------------------------
__global__ void k_stats(const float* __restrict__ adj, float* __restrict__ ws) {
  int t = blockIdx.x * blockDim.x + threadIdx.x;   // 0..B*N*H-1
  int h = t & 7;
  int i = (t >> 3) & (N_ - 1);
  int b = t >> 13;
  float esrc = ws[WS_ESRC + ((b * N_ + i) << 3) + h];
  const float* ed   = ws + WS_EDST + ((size_t)b * N_ << 3);
  const float* arow = adj + (size_t)i * N_;

  float mmax = -1e30f;
  for (int j = 0; j < N_; ++j) {
    float v = esrc + ed[j * 8 + h];
    v = v > 0.f ? v : 0.2f * v;                 // leaky_relu(0.2)
    v = (arow[j] == 0.f) ? -1.0e9f : v;         // adjacency mask
    mmax = fmaxf(mmax, v);
  }
  float ssum = 0.f;
  for (int j = 0; j < N_; ++j) {
    float v = esrc + ed[j * 8 + h];
    v = v > 0.f ? v : 0.2f * v;
    v = (arow[j] == 0.f) ? -1.0e9f : v;
    ssum += __expf(v - mmax);
  }
  ws[WS_M + t] = mmax;
  ws[WS_S + t] = ssum;
}

// ---------------------------------------------------------------------------
// Kernel 3: lap_out = L @ hw + bias via fp32 WMMA (v_wmma_f32_16x16x4_f32).
// Wave-per-16x16 tile; 1024 waves (B=4 x 64 i-tiles x 4 k-tiles); K loop 1024/4.
// Output goes to out[..., 64:128].
// ---------------------------------------------------------------------------
__global__ void k_lap(const float* __restrict__ L, const float* __restrict__ bias,
                      const float* __restrict__ ws, float* __restrict__ out) {
  int lane = threadIdx.x & 31;
  int wave = blockIdx.x * (blockDim.x >> 5) + (threadIdx.x >> 5);  // 0..1023
  int k0 = (wave & 3) << 4;
  int i0 = ((wave >> 2) & 63) << 4;
  int b  = wave >> 8;

  const float* hw = ws + WS_HW + (size_t)b * N_ * OD;
  int m    = lane & 15;          // A: row M; B/C: col N
  int koff = (lane >> 4) << 1;   // upper half-wave holds K=2,3
  const float* Lr = L + (size_t)(i0 + m) * N_;

  v8f c = {};
  for (int j0 = 0; j0 < N_; j0 += 4) {
    if ((j0 & 63) == 0)
      __builtin_prefetch(Lr + j0 + 64, 0, 1);          // global_prefetch_b8
    v2f a, bb;
    a.x  = Lr[j0 + koff];
    a.y  = Lr[j0 + koff + 1];
    bb.x = hw[(j0 + koff)     * OD + k0 + m];
    bb.y = hw[(j0 + koff + 1) * OD + k0 + m];
    c = __builtin_amdgcn_wmma_f32_16x16x4_f32(false, a, false, bb,
                                              (short)0, c, false, false);
  }
  int half = (lane >> 4) << 3;   // upper lanes hold rows M+8
#pragma unroll
  for (int r = 0; r < 8; ++r) {
    int i = i0 + r + half;
    out[((size_t)b * N_ + i) * 128 + 64 + k0 + m] = c[r] + bias[k0 + m];
  }
}

// ---------------------------------------------------------------------------
// Kernel 4: gat_out[b,i,h,:] = softmax-weighted sum of ht over j, via WMMA.
// Wave per (b, 16-row i-tile, head): A = exp(masked lrelu logits - m) built
// with branch-free selects (EXEC stays all-1s for WMMA), B = ht columns
// (8 of 16 N-columns used). 2048 waves x 256 WMMAs.
// ---------------------------------------------------------------------------
__global__ void k_gat(const float* __restrict__ adj, const float* __restrict__ ws,
                      float* __restrict__ out) {
  int lane = threadIdx.x & 31;
  int wave = blockIdx.x * (blockDim.x >> 5) + (threadIdx.x >> 5);  // 0..2047
  int h  = wave & 7;
  int i0 = ((wave >> 3) & 63) << 4;
  int b  = wave >> 9;

  int m    = lane & 15;
  int koff = (lane >> 4) << 1;
  int rowi = i0 + m;
  const float* arow = adj + (size_t)rowi * N_;
  float esrc = ws[WS_ESRC + ((size_t)(b * N_ + rowi) << 3) + h];
  float mrow = ws[WS_M    + ((size_t)(b * N_ + rowi) << 3) + h];
  const float* ed = ws + WS_EDST + ((size_t)b * N_ << 3);
  const float* ht = ws + WS_HT   + (size_t)b * N_ * OD;
  bool nvalid = m < 8;           // only 8 of 16 B-matrix columns carry data

  v8f c = {};
  for (int j0 = 0; j0 < N_; j0 += 4) {
    int jx = j0 + koff, jy = jx + 1;
    float vx = esrc + ed[jx * 8 + h];
    vx = vx > 0.f ? vx : 0.2f * vx;
    vx = (arow[jx] == 0.f) ? -1.0e9f : vx;
    float vy = esrc + ed[jy * 8 + h];
    vy = vy > 0.f ? vy : 0.2f * vy;
    vy = (arow[jy] == 0.f) ? -1.0e9f : vy;
    v2f a, bb;
    a.x = __expf(vx - mrow);
    a.y = __expf(vy - mrow);
    bb.x = nvalid ? ht[jx * OD + h * 8 + m] : 0.f;   // select, not branch
    bb.y = nvalid ? ht[jy * OD + h * 8 + m] : 0.f;
    c = __builtin_amdgcn_wmma_f32_16x16x4_f32(false, a, false, bb,
                                              (short)0, c, false, false);
  }
  int half = (lane >> 4) << 3;
  if (nvalid) {                  // divergence only after all WMMAs are done
#pragma unroll
    for (int r = 0; r < 8; ++r) {
      int i = i0 + r + half;
      float inv = 1.0f / ws[WS_S + ((size_t)(b * N_ + i) << 3) + h];
      out[((size_t)b * N_ + i) * 128 + h * 8 + m] = c[r] * inv;
    }
  }
}

// ---------------------------------------------------------------------------
extern "C" void kernel_launch(void* const* d_in, const int* in_sizes, int n_in,
                              void* d_out, int out_size, void* d_ws, size_t ws_size,
                              hipStream_t stream) {
  const float* inp  = (const float*)d_in[0];  // (4,1024,1)
  const float* hid  = (const float*)d_in[1];  // (4,1024,64)
  const float* W    = (const float*)d_in[2];  // (65,64)
  const float* attn = (const float*)d_in[3];  // (16,1)
  const float* wts  = (const float*)d_in[4];  // (65,64)
  const float* bias = (const float*)d_in[5];  // (64,)
  const float* adj  = (const float*)d_in[6];  // (1024,1024)
  const float* lap  = (const float*)d_in[7];  // (1024,1024)
  float* out = (float*)d_out;                 // (4,1024,128)
  float* ws  = (float*)d_ws;

  k_proj <<<B_ * N_ / 256, 256, 0, stream>>>(inp, hid, W, attn, wts, ws);
  k_stats<<<B_ * N_ * NUM_HEADS / 256, 256, 0, stream>>>(adj, ws);
  k_lap  <<<128, 256, 0, stream>>>(lap, bias, ws, out);   // 1024 waves
  k_gat  <<<256, 256, 0, stream>>>(adj, ws, out);         // 2048 waves
}